// GATLayer_43009802502820
// MI455X (gfx1250) — compile-verified
//
#include <hip/hip_runtime.h>
#include <hip/hip_bf16.h>
#include <math.h>

// ---------------------------------------------------------------------------
// GAT layer for MI455X (gfx1250, wave32).
//   h      = x @ W            (N x 256) x (256 x 64, head-major)  -> WMMA f32
//   e_s/e_d per (node, head)  tiny dot products
//   edge softmax over dst segments (atomic max / exp / atomic sum)
//   h_out[dst] += alpha * h[src]   (atomic f32 adds), alpha_avg per edge
// Workspace layout (floats): h[N*64] | es[N*8] | ed[N*8] | emax[N*8] | esum[N*8]
//   = 9.6M floats = 38.4 MB.
// ---------------------------------------------------------------------------

typedef __attribute__((ext_vector_type(2))) float v2f;
typedef __attribute__((ext_vector_type(8))) float v8f;

#define IN_DIM    256
#define OUT_COLS  64      // NUM_HEADS * HEAD_DIM
#define NUM_HEADS 8
#define HEAD_DIM  8
#define NEG_SLOPE 0.2f
#define GAT_EPS   1e-16f
#define BLK_ROWS  32

// ---------------------------------------------------------------------------
// Kernel 1: h = x @ Wcat via V_WMMA_F32_16X16X4_F32.
// Block = 256 threads = 8 waves; block computes 32 rows x 64 cols.
// Wave w: row-tile = w>>2, col-tile = w&3 -> one 16x16 output tile per wave,
// accumulating K=256 in steps of 4 (64 WMMA ops per wave).
// ---------------------------------------------------------------------------
__global__ __launch_bounds__(256) void gat_gemm_wmma(
    const float* __restrict__ x,    // [N, 256]
    const float* __restrict__ W,    // [8, 256, 8]
    float* __restrict__ hbuf,       // [N, 64] head-major cols (c = h*8+f)
    int n)
{
    __shared__ float xs[BLK_ROWS][IN_DIM + 1];   // +1 pad: conflict-free col reads
    __shared__ float ws[IN_DIM][OUT_COLS + 1];   // +1 pad

    const int tid     = threadIdx.x;
    const int rowbase = blockIdx.x * BLK_ROWS;

    // Stage Wcat into LDS: ws[d][h*8+f] = W[h*2048 + d*8 + f]; linear reads.
    for (int i = tid; i < IN_DIM * OUT_COLS; i += 256) {
        int hh = i >> 11;          // / 2048
        int d  = (i & 2047) >> 3;  // dim index
        int f  = i & 7;
        ws[d][hh * 8 + f] = W[i];
    }
    // Stage x tile (coalesced; clamp row for safety, N%32==0 in practice).
    for (int i = tid; i < BLK_ROWS * IN_DIM; i += 256) {
        int r  = i >> 8;
        int c  = i & 255;
        int rr = rowbase + r;
        if (rr >= n) rr = n - 1;
        xs[r][c] = x[(size_t)rr * IN_DIM + c];
    }
    __syncthreads();

    const int wave  = tid >> 5;
    const int lane  = tid & 31;
    const int rt    = wave >> 2;          // 0..1
    const int ct    = wave & 3;           // 0..3
    const int lrow  = lane & 15;
    const int khalf = (lane >> 4) << 1;   // lanes 0-15 -> K=0,1 ; 16-31 -> K=2,3
    const int col   = ct * 16 + lrow;     // B/C column for this lane
    const int arow  = rt * 16 + lrow;     // A row for this lane

    v8f acc = {};
    #pragma unroll 8
    for (int k = 0; k < IN_DIM; k += 4) {
        v2f a, b;
        a.x = xs[arow][k + khalf];
        a.y = xs[arow][k + khalf + 1];
        b.x = ws[k + khalf][col];
        b.y = ws[k + khalf + 1][col];
        // D = A(16x4) * B(4x16) + C ; 8 args: neg_a, A, neg_b, B, c_mod, C, reuse_a, reuse_b
        acc = __builtin_amdgcn_wmma_f32_16x16x4_f32(
            false, a, false, b, (short)0, acc, false, false);
    }

    // C/D layout: VGPR j -> M = 8*(lane>>4) + j, N = lane&15 (+16*ct).
    const int mbase = (lane >> 4) * 8;
    #pragma unroll
    for (int j = 0; j < 8; ++j) {
        int row = rowbase + rt * 16 + mbase + j;
        if (row < n)
            hbuf[(size_t)row * OUT_COLS + col] = acc[j];
    }
}

// ---------------------------------------------------------------------------
// Kernel 2: per (node, head) attention logits  es = h.a_src, ed = h.a_dst
// ---------------------------------------------------------------------------
__global__ void gat_esd(const float* __restrict__ hbuf,
                        const float* __restrict__ a,   // [8, 16]
                        float* __restrict__ es, float* __restrict__ ed, int n)
{
    int tid = blockIdx.x * blockDim.x + threadIdx.x;
    if (tid >= n * NUM_HEADS) return;
    int node = tid >> 3;
    int hh   = tid & 7;
    const float* hp = hbuf + (size_t)node * OUT_COLS + hh * HEAD_DIM;
    const float* ap = a + hh * (2 * HEAD_DIM);
    float s = 0.f, d = 0.f;
    #pragma unroll
    for (int f = 0; f < HEAD_DIM; ++f) {
        float v = hp[f];
        s += v * ap[f];
        d += v * ap[HEAD_DIM + f];
    }
    es[tid] = s;
    ed[tid] = d;
}

// ---------------------------------------------------------------------------
// Kernel 3: init emax=-inf, esum=0, h_out=0
// ---------------------------------------------------------------------------
__global__ void gat_init(float* __restrict__ hout, float* __restrict__ emax,
                         float* __restrict__ esum, int n)
{
    int tid = blockIdx.x * blockDim.x + threadIdx.x;
    if (tid < n * OUT_COLS) hout[tid] = 0.f;
    if (tid < n * NUM_HEADS) {
        emax[tid] = -INFINITY;
        esum[tid] = 0.f;
    }
}

__device__ __forceinline__ float gat_logit(const float* es, const float* ed,
                                           long long s, long long d, int hh)
{
    float v = es[s * NUM_HEADS + hh] + ed[d * NUM_HEADS + hh];
    return v > 0.f ? v : NEG_SLOPE * v;
}

// ---------------------------------------------------------------------------
// Kernel 4: segment max over dst (float atomic max via int/uint trick)
// ---------------------------------------------------------------------------
__global__ void gat_edge_max(const long long* __restrict__ ei,
                             const float* __restrict__ es,
                             const float* __restrict__ ed,
                             float* __restrict__ emax, long long E)
{
    long long tid = (long long)blockIdx.x * blockDim.x + threadIdx.x;
    if (tid >= E * NUM_HEADS) return;
    long long e = tid >> 3;
    int hh = (int)(tid & 7);
    long long s = ei[e], d = ei[E + e];
    float v = gat_logit(es, ed, s, d, hh);
    float* addr = &emax[d * NUM_HEADS + hh];
    if (v >= 0.f)
        atomicMax((int*)addr, __float_as_int(v));
    else
        atomicMin((unsigned int*)addr, __float_as_uint(v));
}

// ---------------------------------------------------------------------------
// Kernel 5: segment sum of exp(e - max)
// ---------------------------------------------------------------------------
__global__ void gat_edge_sum(const long long* __restrict__ ei,
                             const float* __restrict__ es,
                             const float* __restrict__ ed,
                             const float* __restrict__ emax,
                             float* __restrict__ esum, long long E)
{
    long long tid = (long long)blockIdx.x * blockDim.x + threadIdx.x;
    if (tid >= E * NUM_HEADS) return;
    long long e = tid >> 3;
    int hh = (int)(tid & 7);
    long long s = ei[e], d = ei[E + e];
    float v  = gat_logit(es, ed, s, d, hh);
    float ex = expf(v - emax[d * NUM_HEADS + hh]);
    atomicAdd(&esum[d * NUM_HEADS + hh], ex);
}

// ---------------------------------------------------------------------------
// Kernel 6: alpha = ex / (sum+eps); h_out[dst] += alpha*h[src]; alpha_avg.
// 8 consecutive lanes = 8 heads of one edge (4 edges per wave32);
// cross-head mean via __shfl_xor within width-8 groups.
// ---------------------------------------------------------------------------
__global__ void gat_edge_apply(const long long* __restrict__ ei,
                               const float* __restrict__ es,
                               const float* __restrict__ ed,
                               const float* __restrict__ emax,
                               const float* __restrict__ esum,
                               const float* __restrict__ hbuf,
                               float* __restrict__ hout,
                               float* __restrict__ alpha_avg, long long E)
{
    long long tid = (long long)blockIdx.x * blockDim.x + threadIdx.x;
    if (tid >= E * NUM_HEADS) return;
    long long e = tid >> 3;
    int hh = (int)(tid & 7);
    long long s = ei[e], d = ei[E + e];
    float v     = gat_logit(es, ed, s, d, hh);
    float ex    = expf(v - emax[d * NUM_HEADS + hh]);
    float alpha = ex / (esum[d * NUM_HEADS + hh] + GAT_EPS);

    float asum = alpha;
    asum += __shfl_xor(asum, 4, 8);
    asum += __shfl_xor(asum, 2, 8);
    asum += __shfl_xor(asum, 1, 8);
    if (hh == 0) alpha_avg[e] = asum * (1.0f / NUM_HEADS);

    const float* hs = hbuf + (size_t)s * OUT_COLS + hh * HEAD_DIM;
    float*       ho = hout + (size_t)d * OUT_COLS + hh * HEAD_DIM;
    #pragma unroll
    for (int f = 0; f < HEAD_DIM; ++f)
        atomicAdd(&ho[f], alpha * hs[f]);
}

// ---------------------------------------------------------------------------
extern "C" void kernel_launch(void* const* d_in, const int* in_sizes, int n_in,
                              void* d_out, int out_size, void* d_ws, size_t ws_size,
                              hipStream_t stream)
{
    const float*     x  = (const float*)d_in[0];
    const long long* ei = (const long long*)d_in[1];   // int64 [2, E]
    const float*     W  = (const float*)d_in[2];       // [8, 256, 8]
    const float*     a  = (const float*)d_in[3];       // [8, 16]

    const int       N = in_sizes[0] / IN_DIM;
    const long long E = in_sizes[1] / 2;

    float* hbuf = (float*)d_ws;
    float* es   = hbuf + (size_t)N * OUT_COLS;
    float* ed   = es   + (size_t)N * NUM_HEADS;
    float* emax = ed   + (size_t)N * NUM_HEADS;
    float* esum = emax + (size_t)N * NUM_HEADS;

    float* hout      = (float*)d_out;                  // [N, 64]
    float* alpha_avg = hout + (size_t)N * OUT_COLS;    // [E]

    const int threads = 256;

    // 1) projection (WMMA)
    int gemm_blocks = (N + BLK_ROWS - 1) / BLK_ROWS;
    gat_gemm_wmma<<<gemm_blocks, threads, 0, stream>>>(x, W, hbuf, N);

    // 2) per-node logits
    int esd_blocks = (N * NUM_HEADS + threads - 1) / threads;
    gat_esd<<<esd_blocks, threads, 0, stream>>>(hbuf, a, es, ed, N);

    // 3) init accumulators + output
    int init_blocks = (N * OUT_COLS + threads - 1) / threads;
    gat_init<<<init_blocks, threads, 0, stream>>>(hout, emax, esum, N);

    // 4-6) edge passes
    long long ework = E * NUM_HEADS;
    int edge_blocks = (int)((ework + threads - 1) / threads);
    gat_edge_max<<<edge_blocks, threads, 0, stream>>>(ei, es, ed, emax, E);
    gat_edge_sum<<<edge_blocks, threads, 0, stream>>>(ei, es, ed, emax, esum, E);
    gat_edge_apply<<<edge_blocks, threads, 0, stream>>>(ei, es, ed, emax, esum,
                                                        hbuf, hout, alpha_avg, E);
}